// LogicalGNNLayerComplEx_34514357190803
// MI455X (gfx1250) — compile-verified
//
#include <hip/hip_runtime.h>

// ---------------------------------------------------------------------------
// MI455X (gfx1250) LogicalGNNLayerComplEx.
//
// Roofline: message passing ~0.9 GB (memory bound ~40us @ 23.3 TB/s).
// MLP = 67 GFLOP over ~400 MB HBM -> memory bound; FP32 WMMA 16x16x4 keeps
// full f32 accuracy and the matrix pipes are nowhere near the limit.
// 4-wave workgroups (64 rows) stage B tiles into LDS with the Tensor Data
// Mover (tensor_load_to_lds, TENSORcnt, double-buffered) so all 4 waves
// share each B tile: ~2 GB L2 traffic (vs ~8 GB unshared) and one TDM
// instruction per 10 KB (W1) / 4 KB (W2) tile instead of hundreds of b32 loads.
// TDM pad_enable gives LDS row stride 144/80 floats -> conflict-free B reads.
// ---------------------------------------------------------------------------

typedef __attribute__((ext_vector_type(2))) float v2f;
typedef __attribute__((ext_vector_type(8))) float v8f;
typedef __attribute__((ext_vector_type(4))) unsigned int v4u;
typedef __attribute__((ext_vector_type(8))) int v8i_;
typedef __attribute__((ext_vector_type(4))) int v4i_;

#define N_TERMS 16
#define N_EDGES 32
#define BATCH   2048
#define DHALF   500
#define FDIM    1000                 // 2*EMB
#define HIDDEN  512
#define EPS_F   0.1f
#define MROWS   (N_TERMS * BATCH)    // 32768 flattened GEMM rows
#define LDSTR   (HIDDEN + 2)         // padded hdn LDS row stride

#define KSTEP1  20                   // stage-1 K per TDM tile (1000 = 50*20)
#define NKB1    (FDIM / KSTEP1)      // 50
#define BSTR1   (128 + 16)           // LDS row stride of staged W1 tile
#define BUF1    (KSTEP1 * BSTR1)     // 2880 floats per buffer

#define KSTEP2  16                   // stage-2 K per TDM tile (512 = 32*16)
#define NKB2    (HIDDEN / KSTEP2)    // 32
#define BSTR2   (64 + 16)            // LDS row stride of staged W2 tile

// ---------------------------------------------------------------------------
// TDM: 2D tile load Global->LDS. D# built per CDNA5 ISA 8.3-8.6.
// data_size=4B; pad_enable inserts (pad_amount_code+1) DWORDs after every
// 2^(pad_interval_code+1) DWORDs (set = one tile row) -> padded LDS rows.
// ---------------------------------------------------------------------------
__device__ __forceinline__ void tdm_load_2d(
    unsigned lds_off, const void* gptr,
    unsigned tensor_d0, unsigned tensor_d1, unsigned d0_stride,
    unsigned tile_d0, unsigned tile_d1,
    unsigned pad_interval_code, unsigned pad_amount_code)
{
    unsigned long long ga = (unsigned long long)(uintptr_t)gptr;
    v4u g0;
    g0.x = 1u;                                        // count=1, user D#
    g0.y = lds_off;                                   // LDS byte address
    g0.z = (unsigned)ga;                              // global_addr[31:0]
    g0.w = (unsigned)((ga >> 32) & 0x1FFFFFFu)        // global_addr[56:32]
         | (2u << 30);                                // type=2 ("image")
    v8i_ g1;
    g1[0] = (int)((2u << 16)                          // data_size = 4B
                | (1u << 20)                          // pad_enable
                | (pad_interval_code << 22)
                | (pad_amount_code << 25));
    g1[1] = (int)((tensor_d0 & 0xFFFFu) << 16);       // tensor_dim0[15:0]
    g1[2] = (int)((tensor_d0 >> 16) | ((tensor_d1 & 0xFFFFu) << 16));
    g1[3] = (int)((tensor_d1 >> 16) | (tile_d0 << 16));
    g1[4] = (int)(tile_d1 & 0xFFFFu);                 // tile_dim1 (tile_dim2=0)
    g1[5] = (int)d0_stride;                           // tensor_dim0_stride lo32
    g1[6] = 0;
    g1[7] = 0;
    v4i_ zz = {0, 0, 0, 0};                           // 2D: groups 2/3 unused
    v8i_ z8 = {0, 0, 0, 0, 0, 0, 0, 0};               // extra group (unused)
    __builtin_amdgcn_tensor_load_to_lds(g0, g1, zz, zz, z8, 0);
}

// ---------------------------------------------------------------------------
// Kernel 1: agg[n] = EPS*term[n] + ComplEx messages addressed to n.
// Gather formulation (deterministic, no atomics); edge metadata is
// wave-uniform -> scalar loads + scalar branches.
// ---------------------------------------------------------------------------
__global__ __launch_bounds__(256) void agg_kernel(
    const float* __restrict__ term, const float* __restrict__ pred,
    const int* __restrict__ hidx, const int* __restrict__ tidx,
    const int* __restrict__ sbits, float* __restrict__ agg)
{
    const int n    = blockIdx.y;
    const int flat = blockIdx.x * 256 + threadIdx.x;     // < BATCH*DHALF
    const int b    = flat / DHALF;
    const int j    = flat - b * DHALF;
    const size_t base = (size_t)b * FDIM + j;
    const size_t nOff = (size_t)n * BATCH * FDIM;

    float acc0 = EPS_F * term[nOff + base];
    float acc1 = EPS_F * term[nOff + base + DHALF];

    #pragma unroll 1
    for (int e = 0; e < N_EDGES; ++e) {
        const int   he = hidx[e];
        const int   te = tidx[e];
        const float s  = 1.0f - 2.0f * (float)sbits[e];
        const float* pe = pred + (size_t)e * BATCH * FDIM;
        if (he == n) {                                   // msg_head (tail emb)
            const float* tt = term + (size_t)te * BATCH * FDIM;
            const float t0 = tt[base], t1 = tt[base + DHALF];
            const float p0 = pe[base], p1 = pe[base + DHALF];
            acc0 += s * (t0 * p0 + t1 * p1);
            acc1 += s * (t1 * p0 - t0 * p1);
        }
        if (te == n) {                                   // msg_tail (head emb)
            const float* hh = term + (size_t)he * BATCH * FDIM;
            const float h0 = hh[base], h1 = hh[base + DHALF];
            const float p0 = pe[base], p1 = pe[base + DHALF];
            acc0 += s * (h0 * p0 - h1 * p1);
            acc1 += s * (h0 * p1 + h1 * p0);
        }
    }
    agg[nOff + base]         = acc0;
    agg[nOff + base + DHALF] = acc1;
}

// ---------------------------------------------------------------------------
// Kernel 2: out = relu(agg @ W1 + b1) @ W2 + b2, fused per 64-row block.
// 4 waves x 16 rows. B tiles TDM-staged into LDS, double-buffered; hdn tile
// lives in LDS between the two GEMMs (D-layout -> A-layout transpose).
// FP32 WMMA operand layout (16x16x4):
//   A lanes 0-15: M=lane, K={k,k+1};  lanes 16-31: M=lane-16, K={k+2,k+3}
//   B lanes 0-15: N=lane, K={k,k+1};  lanes 16-31: N=lane-16, K={k+2,k+3}
//   C/D vgpr v: lanes 0-15 -> (M=v, N=lane); lanes 16-31 -> (M=v+8, N=lane-16)
// ---------------------------------------------------------------------------
__global__ __launch_bounds__(128) void fused_mlp_kernel(
    const float* __restrict__ agg, const float* __restrict__ W1,
    const float* __restrict__ b1,  const float* __restrict__ W2,
    const float* __restrict__ b2,  float* __restrict__ out)
{
    __shared__ float smem[64 * LDSTR + 2 * BUF1];        // hdn | B staging (151KB)
    float* hdn = smem;
    float* stg = smem + 64 * LDSTR;

    const int lane   = threadIdx.x & 31;
    const int waveId = threadIdx.x >> 5;                 // 0..3
    const int lo     = lane & 15;
    const int hi     = lane >> 4;
    const int rowBase = blockIdx.x * 64;
    const int wrow    = waveId * 16;                     // wave's first local row
    const bool issuer = (threadIdx.x < 32);              // wave 0 drives TDM

    const unsigned stgOff0 = (unsigned)(uintptr_t)(stg);
    const unsigned stgOff1 = (unsigned)(uintptr_t)(stg + BUF1);

    const float* Arow = agg + (size_t)(rowBase + wrow + lo) * FDIM;

    // ---- Stage 1: hdn = relu(agg @ W1 + b1) ------------------------------
    for (int nc = 0; nc < HIDDEN; nc += 128) {           // 4 column chunks
        if (issuer)                                      // prime buffer 0
            tdm_load_2d(stgOff0, W1 + nc, HIDDEN, FDIM, HIDDEN,
                        128, KSTEP1, /*interval 128DW*/6, /*pad 16DW*/15);
        v8f acc[8] = {};
        for (int kb = 0; kb < NKB1; ++kb) {
            if (issuer) {
                if (kb + 1 < NKB1) {
                    tdm_load_2d((kb & 1) ? stgOff0 : stgOff1,
                                W1 + (size_t)(kb + 1) * KSTEP1 * HIDDEN + nc,
                                HIDDEN, FDIM, HIDDEN, 128, KSTEP1, 6, 15);
                    __builtin_amdgcn_s_wait_tensorcnt(1);
                } else {
                    __builtin_amdgcn_s_wait_tensorcnt(0);
                }
            }
            __syncthreads();                             // tile kb visible
            const float* Bb = stg + (kb & 1) * BUF1;
            const int k0 = kb * KSTEP1;
            #pragma unroll
            for (int kk = 0; kk < KSTEP1; kk += 4) {
                const int ka = kk + 2 * hi;
                const v2f a = *(const v2f*)(Arow + k0 + ka);
                #pragma unroll
                for (int s = 0; s < 8; ++s) {
                    v2f bb;
                    bb.x = Bb[ka * BSTR1 + s * 16 + lo];
                    bb.y = Bb[(ka + 1) * BSTR1 + s * 16 + lo];
                    acc[s] = __builtin_amdgcn_wmma_f32_16x16x4_f32(
                        false, a, false, bb, (short)0, acc[s], false, false);
                }
            }
            __syncthreads();                             // done reading tile kb
        }
        #pragma unroll
        for (int s = 0; s < 8; ++s) {
            const int   n    = nc + s * 16 + lo;
            const float bias = b1[n];
            #pragma unroll
            for (int v = 0; v < 8; ++v) {
                float x = acc[s][v] + bias;
                x = x > 0.0f ? x : 0.0f;                 // relu
                hdn[(wrow + v + 8 * hi) * LDSTR + n] = x;
            }
        }
    }
    __syncthreads();

    // ---- Stage 2 (main): out cols [0,960) via TDM-staged W2 --------------
    for (int nc = 0; nc < 960; nc += 64) {               // 15 full chunks
        if (issuer)
            tdm_load_2d(stgOff0, W2 + nc, FDIM, HIDDEN, FDIM,
                        64, KSTEP2, /*interval 64DW*/5, /*pad 16DW*/15);
        v8f acc[4] = {};
        for (int kb = 0; kb < NKB2; ++kb) {
            if (issuer) {
                if (kb + 1 < NKB2) {
                    tdm_load_2d((kb & 1) ? stgOff0 : stgOff1,
                                W2 + (size_t)(kb + 1) * KSTEP2 * FDIM + nc,
                                FDIM, HIDDEN, FDIM, 64, KSTEP2, 5, 15);
                    __builtin_amdgcn_s_wait_tensorcnt(1);
                } else {
                    __builtin_amdgcn_s_wait_tensorcnt(0);
                }
            }
            __syncthreads();
            const float* Bb = stg + (kb & 1) * BUF1;     // reuse staging region
            const int k0 = kb * KSTEP2;
            #pragma unroll
            for (int kk = 0; kk < KSTEP2; kk += 4) {
                const int ka = kk + 2 * hi;
                const v2f a = *(const v2f*)(&hdn[(wrow + lo) * LDSTR + k0 + ka]);
                #pragma unroll
                for (int s = 0; s < 4; ++s) {
                    v2f bb;
                    bb.x = Bb[ka * BSTR2 + s * 16 + lo];
                    bb.y = Bb[(ka + 1) * BSTR2 + s * 16 + lo];
                    acc[s] = __builtin_amdgcn_wmma_f32_16x16x4_f32(
                        false, a, false, bb, (short)0, acc[s], false, false);
                }
            }
            __syncthreads();
        }
        #pragma unroll
        for (int s = 0; s < 4; ++s) {
            const int   n    = nc + s * 16 + lo;
            const float bias = b2[n];
            #pragma unroll
            for (int v = 0; v < 8; ++v) {
                const int r = rowBase + wrow + v + 8 * hi;
                out[(size_t)r * FDIM + n] = acc[s][v] + bias;
            }
        }
    }

    // ---- Stage 2 (tail): cols [960,1000) with clamped direct loads -------
    {
        const int nc = 960;
        v8f acc[3] = {};
        for (int k = 0; k < HIDDEN; k += 4) {
            const int ka = k + 2 * hi;
            const v2f a = *(const v2f*)(&hdn[(wrow + lo) * LDSTR + ka]);
            #pragma unroll
            for (int s = 0; s < 3; ++s) {
                int n = nc + s * 16 + lo;
                if (n >= FDIM) n = FDIM - 1;             // clamp: cols unused
                const float* Bp = W2 + (size_t)ka * FDIM + n;
                v2f bb; bb.x = Bp[0]; bb.y = Bp[FDIM];
                acc[s] = __builtin_amdgcn_wmma_f32_16x16x4_f32(
                    false, a, false, bb, (short)0, acc[s], false, false);
            }
        }
        #pragma unroll
        for (int s = 0; s < 3; ++s) {
            const int n = nc + s * 16 + lo;
            if (n < FDIM) {
                const float bias = b2[n];
                #pragma unroll
                for (int v = 0; v < 8; ++v) {
                    const int r = rowBase + wrow + v + 8 * hi;
                    out[(size_t)r * FDIM + n] = acc[s][v] + bias;
                }
            }
        }
    }
}

// ---------------------------------------------------------------------------
extern "C" void kernel_launch(void* const* d_in, const int* in_sizes, int n_in,
                              void* d_out, int out_size, void* d_ws, size_t ws_size,
                              hipStream_t stream) {
    (void)in_sizes; (void)n_in; (void)out_size; (void)ws_size;
    const float* term  = (const float*)d_in[0];
    const float* pred  = (const float*)d_in[1];
    const int*   hidx  = (const int*)d_in[2];
    const int*   tidx  = (const int*)d_in[3];
    const int*   sbits = (const int*)d_in[4];
    const float* W1    = (const float*)d_in[5];
    const float* b1    = (const float*)d_in[6];
    const float* W2    = (const float*)d_in[7];
    const float* b2    = (const float*)d_in[8];
    float*       outp  = (float*)d_out;
    float*       agg   = (float*)d_ws;      // 16*2048*1000 f32 = 131 MB scratch

    dim3 g1((BATCH * DHALF) / 256, N_TERMS);            // 4000 x 16 blocks
    agg_kernel<<<g1, 256, 0, stream>>>(term, pred, hidx, tidx, sbits, agg);

    fused_mlp_kernel<<<MROWS / 64, 128, 0, stream>>>(agg, W1, b1, W2, b2, outp);
}